// Masking_14654428414191
// MI455X (gfx1250) — compile-verified
//
#include <hip/hip_runtime.h>
#include <stdint.h>

// ---------------------------------------------------------------------------
// Bottom-k masking via 3-level radix select (11+11+10 bits) on monotone float
// keys, exact tie resolution by lowest index, then a single streaming pass.
//
// MI455X-specific reasoning:
//  * mask_weights (134 MB) < global L2 (192 MB): read mw with REGULAR temporal
//    policy so passes 2..N hit L2; only the used-once streams (w, out) use
//    non-temporal hints so they don't evict the cached mw lines.
//  * Histogram passes stage tiles via double-buffered
//    global_load_async_to_lds_b128 + s_wait_asynccnt (ASYNCcnt-tracked DMA),
//    consume with ds_load_b128, bin with DS atomics.
//  * Tie-resolution passes self-disable (early exit) in the common no-partial-
//    tie case, so their 134 MB scan usually costs one skipped launch.
// ---------------------------------------------------------------------------

#define BINS 2048
#define HIST_BLOCKS 1024
#define STREAM_BLOCKS 2048
#define TPB 256

typedef float __attribute__((ext_vector_type(4))) f4;

struct SelState {
  uint32_t k;           // total elements to zero
  uint32_t remaining;   // rank remaining within current prefix
  uint32_t prefix;      // accumulated key prefix
  uint32_t threshold;   // final threshold key T
  uint32_t tie_rank;    // t = # of ties (key==T) to zero
  uint32_t tie_total;   // c = total # of elements with key==T (from final bin)
  uint32_t need_tie;    // 1 if the tie-gather/select passes must run (t < c)
  uint32_t tie_count;   // atomic counter for tie gather
  uint32_t tie_cutoff;  // exclusive index cutoff among ties
  uint32_t pad[3];
};

// Monotone unsigned key: ascending key order == ascending float order.
__device__ __forceinline__ uint32_t fkey(float f) {
  uint32_t b = __float_as_uint(f);
  uint32_t m = (uint32_t)((int32_t)b >> 31) | 0x80000000u;
  return b ^ m;
}

__global__ void k_init(const int* __restrict__ pct, uint32_t n,
                       SelState* __restrict__ st, uint32_t* __restrict__ bins) {
  for (uint32_t i = threadIdx.x; i < BINS; i += blockDim.x) bins[i] = 0u;
  if (threadIdx.x == 0) {
    double p = (double)(*pct) / 100.0;
    uint32_t k = (uint32_t)((1.0 - p) * (double)n);  // trunc, matches Python int()
    st->k = k;
    st->remaining = k;
    st->prefix = 0u;
    st->threshold = 0u;
    st->tie_rank = 0u;
    st->tie_total = 0u;
    st->need_tie = 0u;
    st->tie_count = 0u;
    st->tie_cutoff = 0u;
  }
}

// Histogram one radix level. Tiles of mask_weights are staged into LDS with
// the CDNA5 async global->LDS DMA path (double-buffered), binned via DS atomics.
// Default (RT) cache policy on the async loads keeps mw resident in L2 for the
// later passes.
template <int LEVEL>
__global__ void k_hist(const float* __restrict__ mw, uint32_t n,
                       const SelState* __restrict__ st,
                       uint32_t* __restrict__ bins) {
  __shared__ uint32_t lhist[BINS];
  __shared__ __align__(16) float tile[2][TPB * 4];

  for (uint32_t i = threadIdx.x; i < BINS; i += blockDim.x) lhist[i] = 0u;
  const uint32_t prefix = st->prefix;
  const uint32_t match_mask =
      (LEVEL == 0) ? 0x00000000u : (LEVEL == 1) ? 0xFFE00000u : 0xFFFFFC00u;
  const int bshift = (LEVEL == 0) ? 21 : (LEVEL == 1) ? 10 : 0;
  const uint32_t bmask = (LEVEL == 2) ? 0x3FFu : 0x7FFu;
  __syncthreads();

  const uint32_t tile_elems = TPB * 4u;  // 1024 floats = 4KB per tile
  const uint32_t ntiles = n / tile_elems;
  const uint32_t lds_a[2] = {
      (uint32_t)(uintptr_t)(&tile[0][0]) + threadIdx.x * 16u,
      (uint32_t)(uintptr_t)(&tile[1][0]) + threadIdx.x * 16u};

  uint32_t cur = 0u;
  uint32_t t0 = blockIdx.x;
  if (t0 < ntiles) {
    uint64_t ga =
        (uint64_t)(uintptr_t)(mw + (size_t)t0 * tile_elems + threadIdx.x * 4u);
    asm volatile("global_load_async_to_lds_b128 %0, %1, off"
                 :: "v"(lds_a[0]), "v"(ga) : "memory");
  }
  for (uint32_t t = t0; t < ntiles; t += gridDim.x) {
    const uint32_t nxt = t + gridDim.x;
    if (nxt < ntiles) {
      uint64_t ga =
          (uint64_t)(uintptr_t)(mw + (size_t)nxt * tile_elems + threadIdx.x * 4u);
      asm volatile("global_load_async_to_lds_b128 %0, %1, off"
                   :: "v"(lds_a[cur ^ 1u]), "v"(ga) : "memory");
      asm volatile("s_wait_asynccnt 1" ::: "memory");  // tile t has landed
    } else {
      asm volatile("s_wait_asynccnt 0" ::: "memory");
    }
    __syncthreads();  // whole tile resident in LDS for all waves
    const f4 tv = *(const f4*)&tile[cur][threadIdx.x * 4u];  // ds_load_b128
#pragma unroll
    for (int j = 0; j < 4; ++j) {
      uint32_t key = fkey(tv[j]);
      if ((key & match_mask) == (prefix & match_mask))
        atomicAdd(&lhist[(key >> bshift) & bmask], 1u);
    }
    __syncthreads();  // protect tile[cur] before it is re-filled
    cur ^= 1u;
  }

  // Tail (n not a multiple of tile size) — plain loads.
  const uint32_t tail_base = ntiles * tile_elems;
  for (uint32_t i = tail_base + blockIdx.x * blockDim.x + threadIdx.x; i < n;
       i += gridDim.x * blockDim.x) {
    uint32_t key = fkey(mw[i]);
    if ((key & match_mask) == (prefix & match_mask))
      atomicAdd(&lhist[(key >> bshift) & bmask], 1u);
  }
  __syncthreads();
  for (uint32_t i = threadIdx.x; i < BINS; i += blockDim.x)
    if (lhist[i]) atomicAdd(&bins[i], lhist[i]);
}

// Tiny single-block scan: find the bin containing the remaining rank,
// fold it into the prefix, and zero bins for the next level.
__global__ void k_scan(SelState* __restrict__ st, uint32_t* __restrict__ bins,
                       int level) {
  if (threadIdx.x == 0) {
    const uint32_t nbins = (level < 2) ? 2048u : 1024u;
    uint32_t rem = st->remaining;
    uint32_t cum = 0u, b = 0u, newrem = rem, cbin = 0u;
    for (uint32_t i = 0; i < nbins; ++i) {
      uint32_t c = bins[i];
      if (cum + c >= rem) { b = i; newrem = rem - cum; cbin = c; break; }
      cum += c;
    }
    uint32_t prefix = st->prefix;
    if (level == 0)      prefix |= (b << 21);
    else if (level == 1) prefix |= (b << 10);
    else                 prefix |= b;
    st->prefix = prefix;
    st->remaining = newrem;
    if (level == 2) {
      st->threshold = prefix;
      st->tie_rank = newrem;
      st->tie_total = cbin;
      if (newrem == cbin) {
        // All ties get zeroed: no gather/select passes needed (common case).
        st->tie_cutoff = 0xFFFFFFFFu;
        st->need_tie = 0u;
      } else {
        st->need_tie = 1u;
      }
    }
  }
  __syncthreads();
  for (uint32_t i = threadIdx.x; i < BINS; i += blockDim.x) bins[i] = 0u;
}

// Gather indices of elements whose key equals the threshold (rare path).
// mw is read with temporal policy: it is L2-resident after the hist passes.
__global__ void k_tiegather(const float* __restrict__ mw, uint32_t n,
                            SelState* __restrict__ st,
                            uint32_t* __restrict__ list, uint32_t cap) {
  if (st->need_tie == 0u) return;  // common case: whole pass skipped
  const uint32_t T = st->threshold;
  const uint32_t stride = gridDim.x * blockDim.x * 4u;
  uint32_t i = (blockIdx.x * blockDim.x + threadIdx.x) * 4u;
  for (; i + 3u < n; i += stride) {
    f4 mv = *(const f4*)(mw + i);  // temporal: expect L2 hit
#pragma unroll
    for (int j = 0; j < 4; ++j) {
      if (fkey(mv[j]) == T) {
        uint32_t s = atomicAdd(&st->tie_count, 1u);
        if (s < cap) list[s] = i + (uint32_t)j;
      }
    }
  }
  for (; i < n; ++i) {
    if (fkey(mw[i]) == T) {
      uint32_t s = atomicAdd(&st->tie_count, 1u);
      if (s < cap) list[s] = i;
    }
  }
}

// Among the m tie indices, find the t-th smallest (O(m^2)/block, m is tiny);
// set exclusive cutoff so exactly t lowest-index ties get zeroed.
__global__ void k_tieselect(SelState* __restrict__ st,
                            const uint32_t* __restrict__ list, uint32_t cap) {
  if (st->need_tie == 0u) return;
  const uint32_t m = st->tie_count;
  const uint32_t t = st->tie_rank;
  if (t == 0u) return;  // cutoff stays 0: no ties zeroed
  if (m > cap) {        // capacity overflow (practically unreachable): zero all ties
    if (threadIdx.x == 0) st->tie_cutoff = 0xFFFFFFFFu;
    return;
  }
  for (uint32_t c = threadIdx.x; c < m; c += blockDim.x) {
    uint32_t v = list[c];
    uint32_t cnt = 0u;
    for (uint32_t j = 0; j < m; ++j) cnt += (list[j] < v) ? 1u : 0u;
    if (cnt == t - 1u) st->tie_cutoff = v + 1u;  // exactly one thread matches
  }
}

// Final streaming pass: out[i] = zero? 0 : w[i].
// w/out are used exactly once -> non-temporal (don't evict mw from L2);
// mw is L2-resident from the hist passes -> regular temporal loads.
__global__ void k_mask(const float* __restrict__ w, const float* __restrict__ mw,
                       float* __restrict__ out, uint32_t n,
                       const SelState* __restrict__ st) {
  const uint32_t T = st->threshold;
  const uint32_t cut = st->tie_cutoff;
  const uint32_t k = st->k;
  const size_t stride = (size_t)gridDim.x * blockDim.x * 8u;
  size_t i = ((size_t)blockIdx.x * blockDim.x + threadIdx.x) * 8u;
  for (; i + 7u < n; i += stride) {
    __builtin_prefetch(w + i + stride, 0, 0);
    f4 wv0 = __builtin_nontemporal_load((const f4*)(w + i));
    f4 wv1 = __builtin_nontemporal_load((const f4*)(w + i + 4));
    f4 mv0 = *(const f4*)(mw + i);      // temporal: L2 hit expected
    f4 mv1 = *(const f4*)(mw + i + 4);
    f4 ov0, ov1;
#pragma unroll
    for (int j = 0; j < 4; ++j) {
      uint32_t key0 = fkey(mv0[j]);
      uint32_t key1 = fkey(mv1[j]);
      bool z0 = (k != 0u) &&
                ((key0 < T) || (key0 == T && (uint32_t)(i + j) < cut));
      bool z1 = (k != 0u) &&
                ((key1 < T) || (key1 == T && (uint32_t)(i + 4 + j) < cut));
      ov0[j] = z0 ? 0.0f : wv0[j];
      ov1[j] = z1 ? 0.0f : wv1[j];
    }
    __builtin_nontemporal_store(ov0, (f4*)(out + i));
    __builtin_nontemporal_store(ov1, (f4*)(out + i + 4));
  }
  for (; i < n; ++i) {
    uint32_t key = fkey(mw[i]);
    bool zero = (k != 0u) && ((key < T) || (key == T && (uint32_t)i < cut));
    out[i] = zero ? 0.0f : w[i];
  }
}

extern "C" void kernel_launch(void* const* d_in, const int* in_sizes, int n_in,
                              void* d_out, int out_size, void* d_ws, size_t ws_size,
                              hipStream_t stream) {
  const float* w  = (const float*)d_in[0];
  const float* mw = (const float*)d_in[1];
  const int* pct  = (const int*)d_in[2];
  float* out = (float*)d_out;
  const uint32_t n = (uint32_t)in_sizes[0];

  SelState* st    = (SelState*)d_ws;
  uint32_t* bins  = (uint32_t*)((char*)d_ws + 64);
  uint32_t* tlist = (uint32_t*)((char*)d_ws + 64 + BINS * 4);
  const size_t reserved = 64 + BINS * 4;
  uint32_t cap = (ws_size > reserved + 16) ? (uint32_t)((ws_size - reserved) / 4)
                                           : 0u;

  k_init<<<1, TPB, 0, stream>>>(pct, n, st, bins);
  k_hist<0><<<HIST_BLOCKS, TPB, 0, stream>>>(mw, n, st, bins);
  k_scan<<<1, TPB, 0, stream>>>(st, bins, 0);
  k_hist<1><<<HIST_BLOCKS, TPB, 0, stream>>>(mw, n, st, bins);
  k_scan<<<1, TPB, 0, stream>>>(st, bins, 1);
  k_hist<2><<<HIST_BLOCKS, TPB, 0, stream>>>(mw, n, st, bins);
  k_scan<<<1, TPB, 0, stream>>>(st, bins, 2);
  k_tiegather<<<STREAM_BLOCKS, TPB, 0, stream>>>(mw, n, st, tlist, cap);
  k_tieselect<<<1, TPB, 0, stream>>>(st, tlist, cap);
  k_mask<<<STREAM_BLOCKS, TPB, 0, stream>>>(w, mw, out, n, st);
}